// RoipoolingLayer_76184129896573
// MI455X (gfx1250) — compile-verified
//
#include <hip/hip_runtime.h>
#include <hip/hip_bf16.h>

#define POOL 7
#define CROP 14
#define H 64
#define W 64
#define C 512
#define C4 (C / 4)          // 128 float4 channel-groups -> one per thread
#define NROI 512

typedef __attribute__((ext_vector_type(4))) float f32x4;

// Compute clipped bilinear tap coordinates exactly like the reference:
//   y0 = clip(floor(s), 0, 63); y1 = clip(y0+1, 0, 63); w = s - floor(s)
__device__ __forceinline__ void tap_coords(float s, int& i0, int& i1, float& w) {
    float f0 = floorf(s);
    w = s - f0;
    int i = (int)f0;
    i0 = min(max(i, 0), H - 1);
    i1 = min(i0 + 1, H - 1);
}

__device__ __forceinline__ float4 lerp4(float4 a, float4 b, float w) {
    // matches reference: a + (b - a) * w  (per component)
    float4 r;
    r.x = a.x + (b.x - a.x) * w;
    r.y = a.y + (b.y - a.y) * w;
    r.z = a.z + (b.z - a.z) * w;
    r.w = a.w + (b.w - a.w) * w;
    return r;
}

__device__ __forceinline__ float4 max4(float4 a, float4 b) {
    float4 r;
    r.x = fmaxf(a.x, b.x);
    r.y = fmaxf(a.y, b.y);
    r.z = fmaxf(a.z, b.z);
    r.w = fmaxf(a.w, b.w);
    return r;
}

// 128-bit non-temporal store (clang builtin needs a native ext-vector type).
__device__ __forceinline__ void nt_store4(float4 v, float4* p) {
    f32x4 t = { v.x, v.y, v.z, v.w };
    __builtin_nontemporal_store(t, (f32x4*)p);
}

// One bilinear sample at (y-row pair, x-col pair) for this lane's 4 channels.
__device__ __forceinline__ float4 bilin(const float4* __restrict__ f, int c4,
                                        int y0, int y1, float wy,
                                        int x0, int x1, float wx) {
    float4 v00 = f[(y0 * W + x0) * C4 + c4];
    float4 v01 = f[(y0 * W + x1) * C4 + c4];
    float4 v10 = f[(y1 * W + x0) * C4 + c4];
    float4 v11 = f[(y1 * W + x1) * C4 + c4];
    float4 top = lerp4(v00, v01, wx);
    float4 bot = lerp4(v10, v11, wx);
    return lerp4(top, bot, wy);
}

__global__ __launch_bounds__(C4) void
RoipoolingLayer_76184129896573_kernel(const float* __restrict__ feat,
                                      const float* __restrict__ rois,
                                      float* __restrict__ out) {
    const int py = blockIdx.x;      // 0..6   pooled row
    const int n  = blockIdx.y;      // 0..511 roi
    const int c4 = threadIdx.x;     // 0..127 float4 channel group

    const float4* __restrict__ f = (const float4*)feat;
    float4* __restrict__ o = (float4*)out;

    // Uniform (blockIdx-only) ROI fetch -> scalarized by compiler (s_load +
    // CDNA5 scalar-FP ops for all coordinate math).
    const float by1 = rois[n * 4 + 0];
    const float bx1 = rois[n * 4 + 1];
    const float by2 = rois[n * 4 + 2];
    const float bx2 = rois[n * 4 + 3];
    const float dy = by2 - by1;
    const float dx = bx2 - bx1;

    // Two crop sample rows for this pooled row: i = 2*py, 2*py+1
    // ys_i = (by1 + dy * (i/13)) * 63   (matches reference arithmetic)
    const float ta = (float)(2 * py)     * (1.0f / (CROP - 1));
    const float tb = (float)(2 * py + 1) * (1.0f / (CROP - 1));
    const float ysa = (by1 + dy * ta) * (float)(H - 1);
    const float ysb = (by1 + dy * tb) * (float)(H - 1);

    int ya0, ya1, yb0, yb1;
    float wya, wyb;
    tap_coords(ysa, ya0, ya1, wya);   // loop-invariant across px
    tap_coords(ysb, yb0, yb1, wyb);

    for (int px = 0; px < POOL; ++px) {
        // Two crop sample cols: j = 2*px, 2*px+1
        const float tc0 = (float)(2 * px)     * (1.0f / (CROP - 1));
        const float tc1 = (float)(2 * px + 1) * (1.0f / (CROP - 1));
        const float xs0 = (bx1 + dx * tc0) * (float)(W - 1);
        const float xs1 = (bx1 + dx * tc1) * (float)(W - 1);

        int x00, x01, x10, x11;
        float wx0, wx1;
        tap_coords(xs0, x00, x01, wx0);
        tap_coords(xs1, x10, x11, wx1);

        // gfx1250 global_prefetch_b8: warm L0 with next iteration's tap
        // columns at the (fixed) four y-rows. Free (no counters).
        if (px < POOL - 1) {
            const float tn = (float)(2 * px + 2) * (1.0f / (CROP - 1));
            const float xsn = (bx1 + dx * tn) * (float)(W - 1);
            int nx0 = min(max((int)floorf(xsn), 0), W - 1);
            __builtin_prefetch((const void*)(f + (ya0 * W + nx0) * C4 + c4), 0, 3);
            __builtin_prefetch((const void*)(f + (ya1 * W + nx0) * C4 + c4), 0, 3);
            __builtin_prefetch((const void*)(f + (yb0 * W + nx0) * C4 + c4), 0, 3);
            __builtin_prefetch((const void*)(f + (yb1 * W + nx0) * C4 + c4), 0, 3);
        }

        // 4 bilinear samples (2x2 pool window), then max.
        float4 s00 = bilin(f, c4, ya0, ya1, wya, x00, x01, wx0);
        float4 s01 = bilin(f, c4, ya0, ya1, wya, x10, x11, wx1);
        float4 s10 = bilin(f, c4, yb0, yb1, wyb, x00, x01, wx0);
        float4 s11 = bilin(f, c4, yb0, yb1, wyb, x10, x11, wx1);

        float4 r = max4(max4(s00, s01), max4(s10, s11));

        // out[n, py, px, :] ; float4-indexed, streamed with NT hint:
        // output is write-once (51 MB), keep feat (8.4 MB, ~16x reuse)
        // resident in the near caches instead.
        nt_store4(r, &o[((n * POOL + py) * POOL + px) * C4 + c4]);
    }
}

extern "C" void kernel_launch(void* const* d_in, const int* in_sizes, int n_in,
                              void* d_out, int out_size, void* d_ws, size_t ws_size,
                              hipStream_t stream) {
    (void)in_sizes; (void)n_in; (void)d_ws; (void)ws_size; (void)out_size;
    const float* feat = (const float*)d_in[0];   // [1,64,64,512] f32
    const float* rois = (const float*)d_in[1];   // [512,4] f32
    float* out = (float*)d_out;                  // [512,7,7,512] f32

    dim3 grid(POOL, NROI);   // (7, 512): one block per (pool_row, roi)
    dim3 block(C4);          // 128 threads = 4 x wave32, one float4 chan-group each
    RoipoolingLayer_76184129896573_kernel<<<grid, block, 0, stream>>>(feat, rois, out);
}